// CenterlineDiceLoss_781684048123
// MI455X (gfx1250) — compile-verified
//
#include <hip/hip_runtime.h>

// ---------------- types ----------------
typedef __attribute__((ext_vector_type(2))) float v2f;
typedef __attribute__((ext_vector_type(8))) float v8f;

// ---------------- problem constants (B=8, C=2, H=W=1024) ----------------
#define IMG_H 1024
#define IMG_W 1024
#define WPR 32                         // 32-bit words per packed row
#define LDS_STRIDE 33                  // padded row stride (conflict-free: gcd(33,64)=1)
#define WORDS_PER_IMG (IMG_H * WPR)    // 32768 words = 128 KB per image
#define N_IMGS 16                      // B*C images per input tensor
#define NPIX_PER_INPUT (N_IMGS * IMG_H * IMG_W)   // 16,777,216
#define PACK_WORDS (NPIX_PER_INPUT / 32)          // 524,288 words (2 MB)
#define CH1_PIX (8 * IMG_H * IMG_W)               // 8,388,608 pixels in channel 1
#define RED_BLOCKS 1024
#define RED_THREADS 256

// ---------------- exact f32 full-wave sum via V_WMMA_F32_16X16X4_F32 ----------------
// A(16x4): vgpr0 = {K0 lanes0-15, K2 lanes16-31} = per-lane value, vgpr1 = 0.
// B(4x16) = ones  =>  D[m][n] = v_m + v_{m+16}  (exact f32 adds).
// c[j] holds rowsum(j) (lanes 0-15) / rowsum(8+j) (lanes 16-31); fold + xor16.
__device__ __forceinline__ float waveSum32(float v) {
    v2f a; a.x = v;    a.y = 0.0f;
    v2f b; b.x = 1.0f; b.y = 1.0f;
    v8f c = {};
    c = __builtin_amdgcn_wmma_f32_16x16x4_f32(false, a, false, b, (short)0, c, false, false);
    float t = c[0] + c[1] + c[2] + c[3] + c[4] + c[5] + c[6] + c[7];
    t += __shfl_xor(t, 16, 32);
    return t;   // full 32-lane sum in every lane
}

// ---------------- carry-save adder helpers (bit-sliced, 32 pixels/op) ----------------
__device__ __forceinline__ unsigned fa_s(unsigned a, unsigned b, unsigned c) { return a ^ b ^ c; }
__device__ __forceinline__ unsigned fa_c(unsigned a, unsigned b, unsigned c) { return (a & b) | ((a ^ b) & c); }

// ---------------- pack: binarize (>0.5) and bit-pack via wave32 ballot ----------------
__global__ __launch_bounds__(256) void packKernel(const float* __restrict__ yp,
                                                  const float* __restrict__ yt,
                                                  unsigned* __restrict__ skp,
                                                  unsigned* __restrict__ skt) {
    int i = blockIdx.x * 256 + threadIdx.x;          // pixel index, bit = i & 31 == lane
    unsigned mp = __builtin_amdgcn_ballot_w32(yp[i] > 0.5f);
    unsigned mt = __builtin_amdgcn_ballot_w32(yt[i] > 0.5f);
    if ((threadIdx.x & 31) == 0) {
        skp[i >> 5] = mp;
        skt[i >> 5] = mt;
    }
}

// ---------------- one Zhang-Suen sub-iteration for one packed row ----------------
// src/dst are LDS buffers (row stride LDS_STRIDE). Returns OR of removed bits.
__device__ __forceinline__ unsigned thinRowPass(const unsigned* src, unsigned* dst,
                                                int r, int step) {
    const int  rowM = r * LDS_STRIDE;
    const bool hasU = (r > 0), hasD = (r < IMG_H - 1);
    const int  rowU = rowM - LDS_STRIDE, rowD = rowM + LDS_STRIDE;

    unsigned pU = 0, pM = 0, pD = 0;                    // previous words (west bits)
    unsigned cU = hasU ? src[rowU] : 0u;
    unsigned cM = src[rowM];
    unsigned cD = hasD ? src[rowD] : 0u;
    unsigned removedAny = 0;

    for (int w = 0; w < WPR; ++w) {
        unsigned nU = 0, nM = 0, nD = 0;                // next words (east bits)
        if (w < WPR - 1) {
            nM = src[rowM + w + 1];
            if (hasU) nU = src[rowU + w + 1];
            if (hasD) nD = src[rowD + w + 1];
        }
        // neighbors (bit b == column w*32+b): P2=N P3=NE P4=E P5=SE P6=S P7=SW P8=W P9=NW
        unsigned P2 = cU, P6 = cD;
        unsigned P4 = (cM >> 1) | (nM << 31), P8 = (cM << 1) | (pM >> 31);
        unsigned P3 = (cU >> 1) | (nU << 31), P9 = (cU << 1) | (pU >> 31);
        unsigned P5 = (cD >> 1) | (nD << 31), P7 = (cD << 1) | (pD >> 31);

        // B = neighbor count, bit-sliced into b0..b3
        unsigned s1 = fa_s(P2, P3, P4), k1 = fa_c(P2, P3, P4);
        unsigned s2 = fa_s(P5, P6, P7), k2 = fa_c(P5, P6, P7);
        unsigned s3 = P8 ^ P9,          k3 = P8 & P9;
        unsigned t1 = fa_s(s1, s2, s3), t2 = fa_c(s1, s2, s3);
        unsigned u1 = fa_s(k1, k2, k3), u2 = fa_c(k1, k2, k3);
        unsigned v1 = t2 ^ u1, v2 = t2 & u1;
        unsigned w1 = u2 ^ v2, w2 = u2 & v2;            // B = t1 +2*v1 +4*w1 +8*w2
        unsigned ge2   = v1 | w1 | w2;                  // B >= 2
        unsigned le6   = ~(w2 | (t1 & v1 & w1));        // B <= 6 (excl. 7,8)
        unsigned condB = ge2 & le6;

        // A = # of 0->1 transitions in cyclic P2..P9,P2 ; need A == 1
        unsigned q0 = ~P2 & P3, q1 = ~P3 & P4, q2 = ~P4 & P5, q3 = ~P5 & P6;
        unsigned q4 = ~P6 & P7, q5 = ~P7 & P8, q6 = ~P8 & P9, q7 = ~P9 & P2;
        unsigned as1 = fa_s(q0, q1, q2), ac1 = fa_c(q0, q1, q2);
        unsigned as2 = fa_s(q3, q4, q5), ac2 = fa_c(q3, q4, q5);
        unsigned as3 = q6 ^ q7,          ac3 = q6 & q7;
        unsigned at1 = fa_s(as1, as2, as3), at2 = fa_c(as1, as2, as3);
        unsigned au1 = fa_s(ac1, ac2, ac3), au2 = fa_c(ac1, ac2, ac3);
        unsigned av1 = at2 ^ au1, av2 = at2 & au1;
        unsigned aw  = au2 | av2;                        // weight-4 bit (A<=4)
        unsigned condA = at1 & ~av1 & ~aw;               // A == 1

        unsigned condC, condD;
        if (step == 0) { condC = ~(P2 & P4 & P6); condD = ~(P4 & P6 & P8); }
        else           { condC = ~(P2 & P4 & P8); condD = ~(P2 & P6 & P8); }

        unsigned rem = cM & condB & condA & condC & condD;
        dst[rowM + w] = cM & ~rem;
        removedAny |= rem;

        pU = cU; pM = cM; pD = cD;
        cU = nU; cM = nM; cD = nD;
    }
    return removedAny;
}

// ---------------- LDS-resident Zhang-Suen thinning to convergence ----------------
// One workgroup per image (32 total). Whole image bit-packed in LDS (2 x ~132 KB
// double buffer fits the 320 KB WGP LDS). Converges in-LDS; HBM touched only at
// load/store of the 128 KB packed image.
__global__ __launch_bounds__(1024) void thinKernel(unsigned* __restrict__ skp,
                                                   unsigned* __restrict__ skt) {
    __shared__ unsigned bufA[IMG_H * LDS_STRIDE];
    __shared__ unsigned bufB[IMG_H * LDS_STRIDE];
    __shared__ int chg;

    const unsigned img = blockIdx.x;
    unsigned* g = (img < N_IMGS) ? (skp + img * WORDS_PER_IMG)
                                 : (skt + (img - N_IMGS) * WORDS_PER_IMG);
    const int tid = threadIdx.x;

    for (int i = tid; i < WORDS_PER_IMG; i += 1024)
        bufA[(i >> 5) * LDS_STRIDE + (i & 31)] = g[i];
    __syncthreads();

    const int r = tid;                                   // one row per thread
    for (int it = 0; it < 128; ++it) {
        if (tid == 0) chg = 0;
        __syncthreads();                                 // reset visible
        unsigned a0 = thinRowPass(bufA, bufB, r, 0);
        if (a0) atomicOr(&chg, 1);
        __syncthreads();                                 // bufB complete
        unsigned a1 = thinRowPass(bufB, bufA, r, 1);
        if (a1) atomicOr(&chg, 1);
        __syncthreads();                                 // bufA complete, flags done
        int changed = chg;
        __syncthreads();                                 // all reads of chg done
        if (!changed) break;                             // uniform exit
    }
    for (int i = tid; i < WORDS_PER_IMG; i += 1024)
        g[i] = bufA[(i >> 5) * LDS_STRIDE + (i & 31)];
}

// ---------------- channel-1 reductions: S1=sum(skp*yt) S2=sum(skp) S3=sum(skt*yp) S4=sum(skt) ----------------
__global__ __launch_bounds__(RED_THREADS) void reduceKernel(const unsigned* __restrict__ skp,
                                                            const unsigned* __restrict__ skt,
                                                            const float* __restrict__ yp,
                                                            const float* __restrict__ yt,
                                                            float* __restrict__ partials) {
    float s1 = 0.0f, s3 = 0.0f;
    int   n2 = 0,    n4 = 0;
    const int stride = RED_BLOCKS * RED_THREADS;         // 262144 divides CH1_PIX exactly
    for (int i = blockIdx.x * RED_THREADS + threadIdx.x; i < CH1_PIX; i += stride) {
        int b    = i >> 20;                              // 1M pixels per channel image
        int off  = i & 0xFFFFF;
        int flat = ((b * 2 + 1) << 20) + off;            // channel 1 of batch b
        unsigned wp = skp[flat >> 5];
        unsigned wt = skt[flat >> 5];
        int bit = flat & 31;
        if ((wp >> bit) & 1u) { s1 += yt[flat]; n2++; }
        if ((wt >> bit) & 1u) { s3 += yp[flat]; n4++; }
    }
    // equal trip counts -> full reconvergence -> EXEC all-ones for WMMA
    float r1 = waveSum32(s1);
    float r2 = waveSum32((float)n2);
    float r3 = waveSum32(s3);
    float r4 = waveSum32((float)n4);

    __shared__ float red[RED_THREADS / 32][4];
    int wid = threadIdx.x >> 5, lane = threadIdx.x & 31;
    if (lane == 0) { red[wid][0] = r1; red[wid][1] = r2; red[wid][2] = r3; red[wid][3] = r4; }
    __syncthreads();
    if (threadIdx.x == 0) {
        float a0 = 0, a1 = 0, a2 = 0, a3 = 0;
        for (int w = 0; w < RED_THREADS / 32; ++w) {
            a0 += red[w][0]; a1 += red[w][1]; a2 += red[w][2]; a3 += red[w][3];
        }
        float* p = partials + (size_t)blockIdx.x * 4;
        p[0] = a0; p[1] = a1; p[2] = a2; p[3] = a3;
    }
}

// ---------------- deterministic final fold + clDice formula ----------------
__global__ __launch_bounds__(32) void finalKernel(const float* __restrict__ partials,
                                                  float* __restrict__ out) {
    int lane = threadIdx.x;                              // single wave, EXEC all-ones
    float s0 = 0, s1 = 0, s2 = 0, s3 = 0;
    for (int i = lane; i < RED_BLOCKS; i += 32) {
        s0 += partials[i * 4 + 0];
        s1 += partials[i * 4 + 1];
        s2 += partials[i * 4 + 2];
        s3 += partials[i * 4 + 3];
    }
    float S1 = waveSum32(s0);
    float S2 = waveSum32(s1);
    float S3 = waveSum32(s2);
    float S4 = waveSum32(s3);
    if (lane == 0) {
        float tprec = (S1 + 1.0f) / (S2 + 1.0f);
        float tsens = (S3 + 1.0f) / (S4 + 1.0f);
        out[0] = 1.0f - 2.0f * (tprec * tsens) / (tprec + tsens);
    }
}

// ---------------- launch ----------------
extern "C" void kernel_launch(void* const* d_in, const int* in_sizes, int n_in,
                              void* d_out, int out_size, void* d_ws, size_t ws_size,
                              hipStream_t stream) {
    const float* yp = (const float*)d_in[0];             // y_pred [8,2,1024,1024] f32
    const float* yt = (const float*)d_in[1];             // y_true [8,2,1024,1024] f32
    float* out = (float*)d_out;                          // scalar f32

    // workspace layout: packed pred (2MB) | packed true (2MB) | block partials (16KB)
    unsigned* skp      = (unsigned*)d_ws;
    unsigned* skt      = skp + PACK_WORDS;
    float*    partials = (float*)(skt + PACK_WORDS);

    packKernel  <<<NPIX_PER_INPUT / 256, 256, 0, stream>>>(yp, yt, skp, skt);
    thinKernel  <<<2 * N_IMGS, 1024, 0, stream>>>(skp, skt);
    reduceKernel<<<RED_BLOCKS, RED_THREADS, 0, stream>>>(skp, skt, yp, yt, partials);
    finalKernel <<<1, 32, 0, stream>>>(partials, out);
}